// CharCNNEmbedding_29944511988350
// MI455X (gfx1250) — compile-verified
//
#include <hip/hip_runtime.h>
#include <cstdint>

#define VOCAB 32000
#define EMB   512
#define KTAPS 3
#define BB    8
#define TT    4096
#define CDIM  (VOCAB * KTAPS)   // 96000 "columns" (v*3+k)
#define RDIM  EMB               // 512 rows (f)

typedef float v4f __attribute__((ext_vector_type(4)));

// ---------------------------------------------------------------------------
// Kernel A: transpose W[f][c] (512 x 96000) -> Tr[c][f] (96000 x 512).
// Tile 32x32 staged through LDS using gfx1250 async global->LDS loads
// (ASYNCcnt path). W is read exactly once -> TH_LOAD_NT so the dead read
// stream does not evict the Tr lines that kernel B will re-read from L2.
// Transposed LDS readback uses stride 33 (coprime with 64 banks) ->
// conflict-free; global stores of Tr are fully coalesced.
// ---------------------------------------------------------------------------
__global__ __launch_bounds__(256) void transpose_w(const float* __restrict__ W,
                                                   float* __restrict__ Tr) {
  __shared__ float tile[32 * 33];
  const int tx = threadIdx.x;          // 0..31 (column within tile)
  const int ty = threadIdx.y;          // 0..7
  const int c0 = blockIdx.x * 32;      // CDIM divisible by 32 (3000 tiles)
  const int f0 = blockIdx.y * 32;      // RDIM divisible by 32 (16 tiles)

  const uint32_t lds_base = (uint32_t)(uintptr_t)(&tile[0]);
  const uint64_t wbase = (uint64_t)(uintptr_t)W;

#pragma unroll
  for (int i = 0; i < 4; ++i) {
    const int fl = ty + i * 8;  // local f row 0..31
    // coalesced 128B/wave global read, lands directly in LDS (no VGPR bounce)
    const uint32_t goff  = ((uint32_t)(f0 + fl) * (uint32_t)CDIM +
                            (uint32_t)(c0 + tx)) * 4u;
    const uint32_t laddr = lds_base + (uint32_t)(fl * 33 + tx) * 4u;
    asm volatile("global_load_async_to_lds_b32 %0, %1, %2 th:TH_LOAD_NT"
                 :
                 : "v"(laddr), "v"(goff), "s"(wbase)
                 : "memory");
  }
  // wait for this wave's async LDS writes, then cross-wave barrier
  asm volatile("s_wait_asynccnt 0x0" ::: "memory");
  __syncthreads();

#pragma unroll
  for (int i = 0; i < 4; ++i) {
    const int cl = ty + i * 8;  // local c row 0..31
    // read transposed: tile[tx*33 + cl] -> bank (33*tx)%64, conflict-free
    Tr[(uint64_t)(c0 + cl) * RDIM + (uint64_t)(f0 + tx)] = tile[tx * 33 + cl];
  }
}

// ---------------------------------------------------------------------------
// Kernel B: 3-tap gather-sum from the transposed table.
// Each 128-lane half-block handles one (b,t) position: 3 contiguous 128-bit
// row reads (table is ~L2-resident) + 1 coalesced 128-bit non-temporal store
// (output is write-once; keep it from evicting the table out of L2).
// ---------------------------------------------------------------------------
__global__ __launch_bounds__(256) void gather_sum(const int* __restrict__ tok,
                                                  const v4f* __restrict__ tr4,
                                                  v4f* __restrict__ out4) {
  const int p  = blockIdx.x * 2 + (threadIdx.x >> 7);  // position in [0, B*T)
  const int f4 = threadIdx.x & 127;                    // v4f index, 128*4=512 f
  const int t  = p & (TT - 1);                         // T is a power of two

  const int tokC = tok[p];
  // center tap: k=1
  v4f a = tr4[(uint64_t)(tokC * 3 + 1) * 128u + (uint64_t)f4];
  if (t > 0) {
    const int tokL = tok[p - 1];  // left tap: k=0
    a += tr4[(uint64_t)(tokL * 3 + 0) * 128u + (uint64_t)f4];
  }
  if (t < TT - 1) {
    const int tokR = tok[p + 1];  // right tap: k=2
    a += tr4[(uint64_t)(tokR * 3 + 2) * 128u + (uint64_t)f4];
  }
  __builtin_nontemporal_store(a, &out4[(uint64_t)p * 128u + (uint64_t)f4]);
}

// ---------------------------------------------------------------------------
// Fallback (workspace too small): direct strided gather from original layout.
// Correct but slow; only used if ws_size < 196.6 MB.
// ---------------------------------------------------------------------------
__global__ __launch_bounds__(512) void gather_direct(const int* __restrict__ tok,
                                                     const float* __restrict__ W,
                                                     float* __restrict__ out) {
  const int p = blockIdx.x;      // position
  const int f = threadIdx.x;     // 0..511
  const int t = p & (TT - 1);
  const float* wf = W + (uint64_t)f * CDIM;
  float s = wf[tok[p] * 3 + 1];
  if (t > 0)      s += wf[tok[p - 1] * 3 + 0];
  if (t < TT - 1) s += wf[tok[p + 1] * 3 + 2];
  __builtin_nontemporal_store(s, &out[(uint64_t)p * EMB + f]);
}

extern "C" void kernel_launch(void* const* d_in, const int* in_sizes, int n_in,
                              void* d_out, int out_size, void* d_ws, size_t ws_size,
                              hipStream_t stream) {
  const int*   tok = (const int*)d_in[0];    // token_ids, B*T elements
  const float* W   = (const float*)d_in[1];  // conv_w, F*V*K floats
  float*       out = (float*)d_out;          // B*T*F floats

  const size_t needed = (size_t)CDIM * (size_t)RDIM * sizeof(float);  // 196.6 MB
  if (ws_size >= needed) {
    float* Tr = (float*)d_ws;
    dim3 blkT(32, 8, 1);
    dim3 grdT(CDIM / 32, RDIM / 32, 1);   // 3000 x 16 tiles
    transpose_w<<<grdT, blkT, 0, stream>>>(W, Tr);
    gather_sum<<<(BB * TT) / 2, 256, 0, stream>>>(tok, (const v4f*)Tr,
                                                  (v4f*)out);
  } else {
    gather_direct<<<BB * TT, 512, 0, stream>>>(tok, W, out);
  }
}